// DigitalFilter_4664334484190
// MI455X (gfx1250) — compile-verified
//
#include <hip/hip_runtime.h>
#include <math.h>

// Problem geometry (fixed by the reference): waveform [32, 2, 480000] f32.
#define T_LEN   480000
#define NCH     64           // 32 * 2 independent sequences
#define SEG     1000         // segments per sequence
#define SEGLEN  480          // samples per segment (30 blocks of 16)
#define BPS     30           // 16-sample WMMA blocks per segment

typedef float v2f __attribute__((ext_vector_type(2)));
typedef float v8f __attribute__((ext_vector_type(8)));

struct BQ { float b0, b1, b2, a1, a2; };   // a0-normalized biquad coefficients

// ---------------------------------------------------------------------------
// Phase 1: per-(channel,segment) zero-state scan -> final DF2T state f = (s1,s2)
// 64000 lanes, fully parallel. Serial depth = 480.
// ---------------------------------------------------------------------------
__global__ __launch_bounds__(256) void bq_seg_state(const float* __restrict__ x,
                                                    float2* __restrict__ f, BQ q)
{
    int idx = blockIdx.x * 256 + threadIdx.x;       // exactly NCH*SEG threads
    int ch  = idx / SEG;
    int seg = idx - ch * SEG;
    const float* p = x + (size_t)ch * T_LEN + (size_t)seg * SEGLEN;
    float s1 = 0.f, s2 = 0.f;
    for (int t = 0; t < SEGLEN; ++t) {
        float xi = p[t];
        float y  = fmaf(q.b0, xi, s1);
        s1 = fmaf(q.b1, xi, fmaf(-q.a1, y, s2));
        s2 = fmaf(q.b2, xi, -q.a2 * y);
    }
    f[idx] = make_float2(s1, s2);
}

// ---------------------------------------------------------------------------
// Phase 2: per-channel serial combine over segments:
//   s_true[k+1] = A^SEGLEN * s_true[k] + f[k],  A = [[-a1, 1], [-a2, 0]]
// Stores the true initial state of every segment. 64 threads, depth 1000.
// ---------------------------------------------------------------------------
__global__ void bq_combine(const float2* __restrict__ f,
                           float2* __restrict__ s0, BQ q)
{
    int ch = threadIdx.x;
    if (ch >= NCH) return;
    float p00 = 1.f, p01 = 0.f, p10 = 0.f, p11 = 1.f;   // P = A^SEGLEN
    for (int i = 0; i < SEGLEN; ++i) {
        float n00 = fmaf(-q.a1, p00, p10);
        float n01 = fmaf(-q.a1, p01, p11);
        float n10 = -q.a2 * p00;
        float n11 = -q.a2 * p01;
        p00 = n00; p01 = n01; p10 = n10; p11 = n11;
    }
    const float2* fc = f  + ch * SEG;
    float2*       sc = s0 + ch * SEG;
    float s1 = 0.f, s2 = 0.f;
    for (int k = 0; k < SEG; ++k) {
        sc[k] = make_float2(s1, s2);
        float2 fk = fc[k];
        float t1 = fmaf(p00, s1, fmaf(p01, s2, fk.x));
        float t2 = fmaf(p10, s1, fmaf(p11, s2, fk.y));
        s1 = t1; s2 = t2;
    }
}

// ---------------------------------------------------------------------------
// Phase 3: WMMA block apply.
// y[0..15] = M(16x20) * [x0..x15, s1, s2, 0, 0] with
//   M[t][k] = h[t-k]   (impulse response, lower-triangular Toeplitz), k<16
//   M[t][16] = g1[t], M[t][17] = g2[t]  (homogeneous responses to s1/s2)
// Five V_WMMA_F32_16X16X4_F32 per 16-sample block; state carried block->block
// from x[14],x[15],y[14],y[15] (exact DF2T identity) via cross-half shuffle.
// Each wave owns 16 (channel,segment) columns; all lanes active (EXEC==~0).
// ---------------------------------------------------------------------------
__device__ __forceinline__ float clamp1(float v) {
    return fminf(fmaxf(v, -1.f), 1.f);
}

__global__ __launch_bounds__(256) void bq_wmma_apply(const float* __restrict__ x,
                                                     float* __restrict__ y,
                                                     const float2* __restrict__ s0,
                                                     BQ q)
{
    __shared__ float sh[48];          // h[16] | g1[16] | g2[16]
    if (threadIdx.x == 0) {
        sh[0]  = q.b0;
        sh[1]  = fmaf(-q.a1, q.b0, q.b1);
        sh[2]  = q.b2 - q.a1 * sh[1] - q.a2 * sh[0];
        sh[16] = 1.f;  sh[17] = -q.a1;
        sh[18] = -q.a1 * sh[17] - q.a2 * sh[16];
        sh[32] = 0.f;  sh[33] = 1.f;
        sh[34] = -q.a1 * sh[33] - q.a2 * sh[32];
        for (int n = 3; n < 16; ++n) {
            sh[n]      = -q.a1 * sh[n - 1]  - q.a2 * sh[n - 2];
            sh[16 + n] = -q.a1 * sh[15 + n] - q.a2 * sh[14 + n];
            sh[32 + n] = -q.a1 * sh[31 + n] - q.a2 * sh[30 + n];
        }
    }
    __syncthreads();
    const float* H  = sh;
    const float* G1 = sh + 16;
    const float* G2 = sh + 32;

    const int lane = threadIdx.x & 31;
    const int r    = lane & 15;               // matrix row (A) / column (B,C,D)
    const bool hi  = lane >= 16;
    const int wave = (blockIdx.x * 256 + threadIdx.x) >> 5;   // 0..3999

    // A operand: lane holds M[r][k0], M[r][k0+1]; low half k0=4j, high k0=4j+2
    v2f A[5];
#pragma unroll
    for (int j = 0; j < 5; ++j) {
        int k0 = 4 * j + (hi ? 2 : 0);
#pragma unroll
        for (int e = 0; e < 2; ++e) {
            int k = k0 + e;
            float v;
            if (k < 16)       v = (k <= r) ? H[r - k] : 0.f;
            else if (k == 16) v = G1[r];
            else if (k == 17) v = G2[r];
            else              v = 0.f;
            if (e == 0) A[j].x = v; else A[j].y = v;
        }
    }

    // This wave's 16 columns: col = wave*16 + r  (lane and lane+16 share a col)
    const int col = wave * 16 + r;
    const int ch  = col / SEG;
    const int seg = col - ch * SEG;
    const size_t base = (size_t)ch * T_LEN + (size_t)seg * SEGLEN;
    const float* xp = x + base;
    float*       yp = y + base;

    float2 st = s0[col];
    float s1 = st.x, s2 = st.y;

    const int off = hi ? 2 : 0;
    for (int blk = 0; blk < BPS; ++blk) {
        const int t0 = blk * 16;
        // B operand rows: low lanes K=4j,4j+1 ; high lanes K=4j+2,4j+3
        float2 xa = *(const float2*)(xp + t0 + 0  + off);
        float2 xb = *(const float2*)(xp + t0 + 4  + off);
        float2 xc = *(const float2*)(xp + t0 + 8  + off);
        float2 xd = *(const float2*)(xp + t0 + 12 + off);
        v2f B0 = {xa.x, xa.y};
        v2f B1 = {xb.x, xb.y};
        v2f B2 = {xc.x, xc.y};
        v2f B3 = {xd.x, xd.y};
        v2f B4 = hi ? (v2f){0.f, 0.f} : (v2f){s1, s2};   // rows 16,17 | 18,19

        v8f c = {};
        c = __builtin_amdgcn_wmma_f32_16x16x4_f32(false, A[0], false, B0, (short)0, c, false, false);
        c = __builtin_amdgcn_wmma_f32_16x16x4_f32(false, A[1], false, B1, (short)0, c, false, false);
        c = __builtin_amdgcn_wmma_f32_16x16x4_f32(false, A[2], false, B2, (short)0, c, false, false);
        c = __builtin_amdgcn_wmma_f32_16x16x4_f32(false, A[3], false, B3, (short)0, c, false, false);
        c = __builtin_amdgcn_wmma_f32_16x16x4_f32(false, A[4], false, B4, (short)0, c, false, false);

        // Next-block state from x[14],x[15],y[14],y[15] (live in high lanes:
        // C VGPR v holds row v+8 there).  DF2T identity:
        //   s2' = b2*x15 - a2*y15
        //   s1' = b1*x15 - a1*y15 + (b2*x14 - a2*y14)
        float x14 = xd.x, x15 = xd.y;
        float y14 = c[6], y15 = c[7];
        float s2n = fmaf(q.b2, x15, -q.a2 * y15);
        float s1n = fmaf(q.b1, x15, fmaf(-q.a1, y15, fmaf(q.b2, x14, -q.a2 * y14)));
        s1 = __shfl(s1n, r + 16, 32);   // low lanes fetch from their high twin
        s2 = __shfl(s2n, r + 16, 32);

        // Store clamped outputs: lane holds rows rb..rb+7 of its column.
        const int rb = hi ? 8 : 0;
        float4 o0 = make_float4(clamp1(c[0]), clamp1(c[1]), clamp1(c[2]), clamp1(c[3]));
        float4 o1 = make_float4(clamp1(c[4]), clamp1(c[5]), clamp1(c[6]), clamp1(c[7]));
        *(float4*)(yp + t0 + rb + 0) = o0;
        *(float4*)(yp + t0 + rb + 4) = o1;
    }
}

// ---------------------------------------------------------------------------
// Host side
// ---------------------------------------------------------------------------
static BQ make_lowpass(double cutoff, double sr) {
    const double Q = 0.7071067811865476;
    double w0 = 2.0 * 3.14159265358979323846 * cutoff / sr;
    double alpha = sin(w0) / (2.0 * Q);
    double cw = cos(w0);
    double b0 = (1.0 - cw) / 2.0, b1 = 1.0 - cw, b2 = b0;
    double a0 = 1.0 + alpha, a1 = -2.0 * cw, a2 = 1.0 - alpha;
    BQ q = { (float)(b0 / a0), (float)(b1 / a0), (float)(b2 / a0),
             (float)(a1 / a0), (float)(a2 / a0) };
    return q;
}

static BQ make_highpass(double cutoff, double sr) {
    const double Q = 0.7071067811865476;
    double w0 = 2.0 * 3.14159265358979323846 * cutoff / sr;
    double alpha = sin(w0) / (2.0 * Q);
    double cw = cos(w0);
    double b0 = (1.0 + cw) / 2.0, b1 = -(1.0 + cw), b2 = b0;
    double a0 = 1.0 + alpha, a1 = -2.0 * cw, a2 = 1.0 - alpha;
    BQ q = { (float)(b0 / a0), (float)(b1 / a0), (float)(b2 / a0),
             (float)(a1 / a0), (float)(a2 / a0) };
    return q;
}

extern "C" void kernel_launch(void* const* d_in, const int* in_sizes, int n_in,
                              void* d_out, int out_size, void* d_ws, size_t ws_size,
                              hipStream_t stream) {
    const float* in  = (const float*)d_in[0];
    float*       out = (float*)d_out;
    float2* f  = (float2*)d_ws;          // NCH*SEG float2  (512 KB)
    float2* s0 = f + (size_t)NCH * SEG;  // NCH*SEG float2  (512 KB)

    BQ lp = make_lowpass(3400.0, 16000.0);
    BQ hp = make_highpass(300.0, 16000.0);

    // Stage 1: lowpass, in -> out (clamped)
    bq_seg_state<<<(NCH * SEG) / 256, 256, 0, stream>>>(in, f, lp);
    bq_combine  <<<1, NCH, 0, stream>>>(f, s0, lp);
    bq_wmma_apply<<<(NCH * SEG) / 16 / 8, 256, 0, stream>>>(in, out, s0, lp);

    // Stage 2: highpass, out -> out in place (clamped)
    bq_seg_state<<<(NCH * SEG) / 256, 256, 0, stream>>>(out, f, hp);
    bq_combine  <<<1, NCH, 0, stream>>>(f, s0, hp);
    bq_wmma_apply<<<(NCH * SEG) / 16 / 8, 256, 0, stream>>>(out, out, s0, hp);
}